// CSPN_5282809774287
// MI455X (gfx1250) — compile-verified
//
#include <hip/hip_runtime.h>

// CSPN 3x3 propagation step, MI455X (gfx1250).
//
// Roofline: 201 MB moved / 23.3 TB/s ~= 8.6 us; 71 MFLOP is noise.
// Pure bandwidth kernel -> the useful CDNA5 paths are data movement:
//   - global_load_async_to_lds_b128 (ASYNCcnt) stages cur_seg rows into LDS
//     (9x reuse across neighboring pixels) without VGPR round-trip.
//   - non-temporal b128 loads for the 151 MB affinity stream (no reuse; keep
//     it out of L2) and non-temporal b128 stores for the output.
// WMMA is intentionally NOT used: weights vary per pixel, so this is an
// elementwise weighted sum, not a GEMM -- matrix ops cannot reduce bytes.

#define IMG_W 512
#define IMG_H 512
#define NBATCH 16
#define HW (IMG_H * IMG_W)
// floats per LDS row: multiple of 4 (keeps 16B alignment of every 4*i slot
// for B128 LDS writes) and !=0 mod 64 banks to stagger rows.
#define LDS_STRIDE 516

typedef float float4v __attribute__((ext_vector_type(4)));

__global__ __launch_bounds__(128) void cspn_step_kernel(
    const float* __restrict__ affinity,   // [16, 9, 512, 512]
    const float* __restrict__ cur_seg,    // [16, 1, 512, 512]
    const float* __restrict__ coarse_seg, // [16, 1, 512, 512]
    float* __restrict__ out)              // [16, 1, 512, 512]
{
    __shared__ __attribute__((aligned(16))) float smem[3 * LDS_STRIDE];

    const int t  = threadIdx.x;       // 0..127
    const int b  = blockIdx.x >> 9;   // blockIdx.x / 512
    const int y  = blockIdx.x & 511;  // row of this block
    const int x0 = t << 2;            // 4 pixels per thread

    // ---- Stage cur_seg rows (y-1, y, y+1) into LDS -------------------------
    // In-bounds rows: one async B128 per lane (128 lanes * 16B = full row).
    // Out-of-image rows: zero-fill (zero padding semantics of the unfold).
    const float* seg_base = cur_seg + (size_t)b * HW;
    const unsigned lds_base = (unsigned)(unsigned long long)(&smem[0]);

#pragma unroll
    for (int r = 0; r < 3; ++r) {
        const int yy = y - 1 + r;
        const unsigned lds_off = lds_base + (unsigned)((r * LDS_STRIDE + x0) * 4);
        if (yy >= 0 && yy < IMG_H) {
            const float* gaddr = seg_base + (size_t)yy * IMG_W + x0;
            asm volatile("global_load_async_to_lds_b128 %0, %1, off"
                         :
                         : "v"(lds_off), "v"(gaddr)
                         : "memory");
        } else {
            float4v z = {0.0f, 0.0f, 0.0f, 0.0f};
            *(float4v*)(&smem[r * LDS_STRIDE + x0]) = z;
        }
    }

    // Each wave drains its own async copies, then cross-wave barrier.
    asm volatile("s_wait_asynccnt 0" ::: "memory");
    __syncthreads();

    // ---- Streamed affinity / coarse loads (non-temporal) -------------------
    const size_t pix   = (size_t)y * IMG_W + x0;
    const float* aff_b = affinity + (size_t)b * 9 * HW + pix;

    float4v a[9];
#pragma unroll
    for (int k = 0; k < 9; ++k)
        a[k] = __builtin_nontemporal_load((const float4v*)(aff_b + (size_t)k * HW));

    const float4v coarse =
        __builtin_nontemporal_load((const float4v*)(coarse_seg + (size_t)b * HW + pix));

    // ---- Gather 3x3 neighborhood from LDS and accumulate -------------------
    float4v acc = {0.0f, 0.0f, 0.0f, 0.0f};

#pragma unroll
    for (int r = 0; r < 3; ++r) {
        const float* srow = &smem[r * LDS_STRIDE];
        const float L  = (x0 == 0)            ? 0.0f : srow[x0 - 1];
        const float c0 = srow[x0 + 0];
        const float c1 = srow[x0 + 1];
        const float c2 = srow[x0 + 2];
        const float c3 = srow[x0 + 3];
        const float R  = (x0 + 4 >= IMG_W)    ? 0.0f : srow[x0 + 4];

        // tap ordering k = 3*ky + kx, ky=0 is row above, kx=0 is left
        const float4v lv = {L,  c0, c1, c2};            // x-1 taps
        const float4v rv = {c1, c2, c3, R};             // x+1 taps
        const float4v cv = (r == 1) ? coarse            // center tap (k=4) replaced
                                    : (float4v){c0, c1, c2, c3};

        acc += a[3 * r + 0] * lv;
        acc += a[3 * r + 1] * cv;
        acc += a[3 * r + 2] * rv;
    }

    __builtin_nontemporal_store(acc, (float4v*)(out + (size_t)b * HW + pix));
}

extern "C" void kernel_launch(void* const* d_in, const int* in_sizes, int n_in,
                              void* d_out, int out_size, void* d_ws, size_t ws_size,
                              hipStream_t stream) {
    const float* affinity   = (const float*)d_in[0];
    const float* cur_seg    = (const float*)d_in[1];
    const float* coarse_seg = (const float*)d_in[2];
    // d_in[3] is the unused python scalar `i`
    float* out = (float*)d_out;

    dim3 grid(NBATCH * IMG_H);  // one block per image row: 16*512 = 8192
    dim3 block(128);            // 128 threads * 4 px = 512 px = full row
    hipLaunchKernelGGL(cspn_step_kernel, grid, block, 0, stream,
                       affinity, cur_seg, coarse_seg, out);
}